// PCA_75307956568420
// MI455X (gfx1250) — compile-verified
//
#include <hip/hip_runtime.h>
#include <stdint.h>

// PCA pipeline for x:[8,4096,1024] fp32, H=16 heads, hs=64.
// K1: column stats (sum/sumsq/nz) + nonzero-row count  (one pass, 128 MB)
// K2: 128x RAW Gram 64x64 (K=4096) via V_WMMA_F32_16X16X4_F32, fed by
//     double-buffered global_load_async_to_lds_b128; centered/normalized
//     Gram recovered by a closed-form rank-1 correction on the 64x64 output.
// K3: 128x symmetric eigh via parallel two-sided Jacobi in LDS.

#define Bc   8
#define Hc   16
#define Sc   4096
#define Dc   1024
#define HSc  64

typedef float v2f __attribute__((ext_vector_type(2)));
typedef float v8f __attribute__((ext_vector_type(8)));

// ---------------------------------------------------------------- Kernel 1
__global__ __launch_bounds__(256) void pca_stats(const float* __restrict__ x,
                                                 float* __restrict__ mean,
                                                 float* __restrict__ invstd,
                                                 float* __restrict__ colsum,
                                                 float* __restrict__ nzrow)
{
    __shared__ unsigned hf[2][8][8];     // [parity][pass][wave] half-row flags
    __shared__ unsigned nzrl[32];
    __shared__ float rs[256], rq[256], rn[256];

    const int t   = threadIdx.x;
    const int gm  = blockIdx.x;            // (b,h), 0..127
    const int b   = gm >> 4, h = gm & 15;
    const int col = t & 63;                // fixed column per thread
    const int rof = t >> 6;                // 0..3
    const size_t xbase = (size_t)b * Sc * Dc + (size_t)h * HSc + col;
    const int wv    = t >> 5;
    const bool lane0 = (t & 31) == 0;

    float sum = 0.f, sq = 0.f, nz = 0.f;
    unsigned nzrcnt = 0;                   // threads t<32 accumulate row flags

    for (int s0 = 0, it = 0; s0 < Sc; s0 += 32, ++it) {
        const int par = it & 1;
        #pragma unroll
        for (int u = 0; u < 8; ++u) {      // 32 rows per barrier
            int row = s0 + 4 * u + rof;
            float v = x[xbase + (size_t)row * Dc];
            sum += v; sq += v * v;
            bool nzb = (v != 0.0f);
            nz += nzb ? 1.0f : 0.0f;
            unsigned m = __builtin_amdgcn_ballot_w32(nzb);
            if (lane0) hf[par][u][wv] = m;
        }
        __syncthreads();
        if (t < 32) {                      // row (s0 + 4u + rr) <- waves 2rr,2rr+1
            int u = t >> 2, rr = t & 3;
            unsigned f = hf[par][u][2*rr] | hf[par][u][2*rr + 1];
            nzrcnt += f ? 1u : 0u;
        }
        // no trailing barrier: next iter writes the other parity
    }

    rs[t] = sum; rq[t] = sq; rn[t] = nz;
    if (t < 32) nzrl[t] = nzrcnt;
    __syncthreads();
    if (t == 0) {
        unsigned a = 0;
        for (int j = 0; j < 32; ++j) a += nzrl[j];
        nzrow[gm] = (float)a;
    }
    if (t < 64) {
        float S = rs[t] + rs[t+64] + rs[t+128] + rs[t+192];
        float Q = rq[t] + rq[t+64] + rq[t+128] + rq[t+192];
        float N = rn[t] + rn[t+64] + rn[t+128] + rn[t+192];
        float mu  = (N != 0.f) ? (S / N) : 0.f;
        float var = (N != 0.f) ? ((Q - 2.f*mu*S + (float)Sc * mu * mu) / N) : 0.f;
        var = fmaxf(var, 0.f);
        float sd  = sqrtf(var);
        mean  [gm * HSc + t] = mu;
        invstd[gm * HSc + t] = (sd != 0.f) ? (1.f / sd) : 0.f;  // div_no_nan
        colsum[gm * HSc + t] = S;
    }
}

// ---------------------------------------------------------------- Kernel 2
#define LP   72                       // LDS row pitch: halves hit disjoint banks
#define BUFB (64 * LP * 4)            // bytes per buffer (18432)

__device__ __forceinline__ void async_fill_b128(unsigned long long sb,
                                                unsigned ldsaddr, unsigned voff)
{
    // GVS mode: mem = SGPR64 + VGPR32 ; vdst = per-lane LDS byte address
    asm volatile("global_load_async_to_lds_b128 %0, %1, %2"
                 :: "v"(ldsaddr), "v"(voff), "s"(sb) : "memory");
}
__device__ __forceinline__ void wait_async0()
{
    asm volatile("s_wait_asynccnt 0x0" ::: "memory");
}

__global__ __launch_bounds__(256) void pca_gram(const float* __restrict__ x,
                                                const float* __restrict__ mean,
                                                const float* __restrict__ invstd,
                                                const float* __restrict__ colsum,
                                                const float* __restrict__ nzrow,
                                                float* __restrict__ G)
{
    __shared__ float lds[2][64 * LP];

    const int t  = threadIdx.x;
    const int gm = blockIdx.x;
    const int b  = gm >> 4, h = gm & 15;
    const float* xblk = x + (size_t)b * Sc * Dc + (size_t)h * HSc;

    // fill assignment: thread -> row = t/16 (+16/pass), 4 cols = (t%16)*4
    const unsigned frow = t >> 4;              // 0..15
    const unsigned fc4  = (t & 15) << 2;       // 0,4,..,60
    const unsigned voff0 = (frow * Dc + fc4) * 4u;            // bytes
    const unsigned lbase = (unsigned)(uintptr_t)&lds[0][0];   // LDS offset
    const unsigned loff0 = lbase + (frow * LP + fc4) * 4u;

    const int wv   = t >> 5;
    const int lane = t & 31;
    const int r    = lane & 15;
    const int half = lane >> 4;
    const int i16  = (wv >> 1) * 16;
    const int j0c  = (wv & 1) * 32;
    const int j1c  = j0c + 16;

    v8f acc0 = {}; v8f acc1 = {};

    // prologue: chunk 0 -> buffer 0
    {
        unsigned long long sb = (unsigned long long)(uintptr_t)xblk;
        #pragma unroll
        for (int p = 0; p < 4; ++p)
            async_fill_b128(sb, loff0 + p * (16u * LP * 4u), voff0 + p * (16u * Dc * 4u));
        wait_async0();
        __syncthreads();
    }

    for (int ch = 0; ch < Sc / 64; ++ch) {
        const int cur = ch & 1;
        if (ch + 1 < Sc / 64) {                // prefetch next chunk -> other buffer
            unsigned long long sb =
                (unsigned long long)(uintptr_t)(xblk + (size_t)(ch + 1) * 64 * Dc);
            unsigned lb = loff0 + (unsigned)(1 - cur) * BUFB;
            #pragma unroll
            for (int p = 0; p < 4; ++p)
                async_fill_b128(sb, lb + p * (16u * LP * 4u), voff0 + p * (16u * Dc * 4u));
        }
        const float* Lb = &lds[cur][0];
        #pragma unroll
        for (int kk = 0; kk < 16; ++kk) {      // 64 rows, K-step = 4
            const int rb = kk * 4 + half * 2;
            const float* L0 = Lb +  rb      * LP;
            const float* L1 = Lb + (rb + 1) * LP;
            v2f a, b0, b1;
            a.x  = L0[i16 + r];  a.y  = L1[i16 + r];
            b0.x = L0[j0c + r];  b0.y = L1[j0c + r];
            b1.x = L0[j1c + r];  b1.y = L1[j1c + r];
            acc0 = __builtin_amdgcn_wmma_f32_16x16x4_f32(
                       false, a, false, b0, (short)0, acc0, false, false);
            acc1 = __builtin_amdgcn_wmma_f32_16x16x4_f32(
                       false, a, false, b1, (short)0, acc1, false, false);
        }
        wait_async0();
        __syncthreads();
    }

    // centered+normalized Gram from raw Gram (rank-1 correction):
    // G = is_I is_J (Graw - mu_I cs_J - mu_J cs_I + mu_I mu_J NZR)
    const float NZ = nzrow[gm];
    const float muJ0 = mean  [gm*HSc + j0c + r];
    const float csJ0 = colsum[gm*HSc + j0c + r];
    const float isJ0 = invstd[gm*HSc + j0c + r];
    const float muJ1 = mean  [gm*HSc + j1c + r];
    const float csJ1 = colsum[gm*HSc + j1c + r];
    const float isJ1 = invstd[gm*HSc + j1c + r];
    float* Gt = G + (size_t)gm * (HSc * HSc);
    #pragma unroll
    for (int vi = 0; vi < 8; ++vi) {
        int I = i16 + vi + 8 * half;           // C/D: VGPR vi -> M = vi + 8*half
        float muI = mean  [gm*HSc + I];
        float csI = colsum[gm*HSc + I];
        float isI = invstd[gm*HSc + I];
        float g0 = (acc0[vi] - muI*csJ0 - muJ0*csI + muI*muJ0*NZ) * isI * isJ0;
        float g1 = (acc1[vi] - muI*csJ1 - muJ1*csI + muI*muJ1*NZ) * isI * isJ1;
        Gt[I * HSc + (j0c + r)] = g0;
        Gt[I * HSc + (j1c + r)] = g1;
    }
}

// ---------------------------------------------------------------- Kernel 3
// Parallel two-sided Jacobi eigh of 64x64 symmetric G; pitch 65 (bank = row+col).
#define GP 65
__global__ __launch_bounds__(256) void pca_eigh(const float* __restrict__ G,
                                                float* __restrict__ out)
{
    __shared__ float Gl[64 * GP];
    __shared__ float Vl[64 * GP];
    __shared__ int   pp[32], qq[32];
    __shared__ float cc[32], ss[32];
    __shared__ float w[64];
    __shared__ int   orig[64];

    const int t  = threadIdx.x;
    const int gm = blockIdx.x;
    const float* Gg = G + (size_t)gm * 4096;

    for (int e = t; e < 4096; e += 256) {
        int row = e >> 6, c = e & 63;
        Gl[row * GP + c] = Gg[e];
        Vl[row * GP + c] = (row == c) ? 1.0f : 0.0f;
    }
    __syncthreads();

    const int k    = t & 31;   // pair id
    const int base = t >> 5;   // 0..7

    for (int sweep = 0; sweep < 12; ++sweep) {
        for (int rr = 0; rr < 63; ++rr) {       // round-robin tournament
            if (t < 32) {
                int j  = t;
                int a  = (j == 0) ? 0 : (1 + (j - 1 + rr) % 63);
                int b2 = 1 + (62 - j + rr) % 63;
                int p = a < b2 ? a : b2;
                int q = a < b2 ? b2 : a;
                pp[t] = p; qq[t] = q;
                float app = Gl[p * GP + p];
                float aqq = Gl[q * GP + q];
                float apq = Gl[p * GP + q];
                float c0 = 1.0f, s0 = 0.0f;
                if (fabsf(apq) > 1e-30f) {
                    float tau = (aqq - app) / (2.0f * apq);
                    float tt  = ((tau >= 0.f) ? 1.0f : -1.0f)
                              / (fabsf(tau) + sqrtf(tau * tau + 1.0f));
                    c0 = 1.0f / sqrtf(tt * tt + 1.0f);
                    s0 = tt * c0;
                }
                cc[t] = c0; ss[t] = s0;
            }
            __syncthreads();
            {   // column update: G <- G*J,  V <- V*J
                int p = pp[k], q = qq[k];
                float c0 = cc[k], s0 = ss[k];
                #pragma unroll
                for (int m = 0; m < 8; ++m) {
                    int i = base + 8 * m;
                    float gp = Gl[i * GP + p], gq = Gl[i * GP + q];
                    Gl[i * GP + p] = c0 * gp - s0 * gq;
                    Gl[i * GP + q] = s0 * gp + c0 * gq;
                    float vp = Vl[i * GP + p], vq = Vl[i * GP + q];
                    Vl[i * GP + p] = c0 * vp - s0 * vq;
                    Vl[i * GP + q] = s0 * vp + c0 * vq;
                }
            }
            __syncthreads();
            {   // row update: G <- J^T * G
                int p = pp[k], q = qq[k];
                float c0 = cc[k], s0 = ss[k];
                #pragma unroll
                for (int m = 0; m < 8; ++m) {
                    int j2 = base + 8 * m;
                    float gp = Gl[p * GP + j2], gq = Gl[q * GP + j2];
                    Gl[p * GP + j2] = c0 * gp - s0 * gq;
                    Gl[q * GP + j2] = s0 * gp + c0 * gq;
                }
            }
            __syncthreads();
        }
    }

    if (t < 64) w[t] = Gl[t * GP + t];
    __syncthreads();
    if (t < 64) {               // ascending rank (eigh order), stable ties
        float wt = w[t];
        int rank = 0;
        for (int j = 0; j < 64; ++j) {
            float wj = w[j];
            if (wj < wt || (wj == wt && j < t)) rank++;
        }
        orig[rank] = t;
    }
    __syncthreads();

    // out[b, h*64 + a, c] = w[orig[c]] * Vl[c][orig[a]] / 8
    float* og = out + (size_t)gm * 4096;
    for (int e = t; e < 4096; e += 256) {
        int a = e >> 6, c = e & 63;
        og[e] = w[orig[c]] * Vl[c * GP + orig[a]] * 0.125f;
    }
}

// ---------------------------------------------------------------- launcher
extern "C" void kernel_launch(void* const* d_in, const int* in_sizes, int n_in,
                              void* d_out, int out_size, void* d_ws, size_t ws_size,
                              hipStream_t stream) {
    (void)in_sizes; (void)n_in; (void)out_size; (void)ws_size;
    const float* x = (const float*)d_in[0];
    float* out = (float*)d_out;

    // workspace (floats): mean | invstd | colsum | nzrow | G   (~2.1 MB)
    float* ws     = (float*)d_ws;
    float* meanp  = ws;                       // 128*64
    float* invstd = meanp  + 128 * HSc;       // 128*64
    float* csum   = invstd + 128 * HSc;       // 128*64
    float* nzrow  = csum   + 128 * HSc;       // 128
    float* G      = nzrow  + 128;             // 128*64*64

    dim3 blk(256);
    pca_stats<<<128, blk, 0, stream>>>(x, meanp, invstd, csum, nzrow);
    pca_gram <<<128, blk, 0, stream>>>(x, meanp, invstd, csum, nzrow, G);
    pca_eigh <<<128, blk, 0, stream>>>(G, out);
}